// MonDEQLayerFC_16750372454624
// MI455X (gfx1250) — compile-verified
//
#include <hip/hip_runtime.h>
#include <math.h>

// MonDEQ layer for MI455X (gfx1250, wave32).
// All GEMMs use V_WMMA_F32_16X16X4_F32 (full fp32 precision, matrix pipe).
// Working set (~7MB) is L2-resident; operands are staged K-major in the
// workspace so every WMMA fragment load is coalesced across lanes.
// Hot-loop latency is hidden with WGP-scope global_prefetch_b8 (inline asm;
// scope 0 pulls L2 lines into the WGP cache) 64 K-rows ahead, issued in a
// branch-free main loop + fully unrolled tail.
// Convergence check is folded into the iterate kernel (deterministic
// fixed-tree reduction, identical in every wave) -> 1 launch per iteration.

typedef __attribute__((ext_vector_type(2))) float v2f;
typedef __attribute__((ext_vector_type(8))) float v8f;

#define N_DIM 1024
#define D_DIM 512
#define B_DIM 256
#define TOL_F 1e-4f
#define NSLOTS 256   // 64 blocks * 4 waves in the iterate grid
#define PF_ROWS 64   // prefetch distance in K-rows (~16 K-steps ahead)

// ------------------------------------------------------------------
// WMMA helper: D = A(16x4,f32) * B(4x16,f32) + C(16x16,f32)
// ------------------------------------------------------------------
__device__ __forceinline__ v8f wmma4f32(v2f a, v2f b, v8f c) {
  // 8 args: (neg_a, A, neg_b, B, c_mod, C, reuse_a, reuse_b)
  return __builtin_amdgcn_wmma_f32_16x16x4_f32(false, a, false, b, (short)0, c,
                                               false, false);
}

__device__ __forceinline__ void zero_acc(v8f acc[2][2]) {
  v8f z = {0.f, 0.f, 0.f, 0.f, 0.f, 0.f, 0.f, 0.f};
  acc[0][0] = z; acc[0][1] = z; acc[1][0] = z; acc[1][1] = z;
}

// WGP-scope prefetch (scope field omitted -> 0 = WGP: fill all levels).
// Base address matches the load base so it CSEs into the same VGPR pair.
#define PF_WGP(ptr, imm_off)                                        \
  asm volatile("global_prefetch_b8 %0, off offset:%c1" ::"v"(ptr), \
               "i"(imm_off))

// One K-step (4 k-values): 8 coalesced b32 loads + 4 WMMAs.
// ap0/bp0 already include the +mt+lm / +nt+lm lane offsets.
template <int LDA, int LDB, bool PF>
__device__ __forceinline__ void gemm_step(const float* __restrict__ ap0,
                                          const float* __restrict__ bp0,
                                          int ka, v8f acc[2][2]) {
  const float* arow = ap0 + (size_t)ka * LDA;
  const float* brow = bp0 + (size_t)ka * LDB;
  if (PF) {
    // Cover all 4 future A rows and all 4 future B rows across lane halves.
    PF_WGP(arow, PF_ROWS * LDA * 4);
    PF_WGP(arow, (PF_ROWS + 1) * LDA * 4);
    PF_WGP(brow, PF_ROWS * LDB * 4);
    PF_WGP(brow, (PF_ROWS + 1) * LDB * 4);
  }
  v2f a0, a1, b0, b1;
  a0.x = arow[0];
  a0.y = arow[LDA];
  a1.x = arow[16];
  a1.y = arow[LDA + 16];
  b0.x = brow[0];
  b0.y = brow[LDB];
  b1.x = brow[16];
  b1.y = brow[LDB + 16];
  acc[0][0] = wmma4f32(a0, b0, acc[0][0]);
  acc[0][1] = wmma4f32(a0, b1, acc[0][1]);
  acc[1][0] = wmma4f32(a1, b0, acc[1][0]);
  acc[1][1] = wmma4f32(a1, b1, acc[1][1]);
}

// ------------------------------------------------------------------
// 32x32 output tile per wave. Operands are K-major:
//   A-op element (M=m,K=k) = Acol[k*LDA + m]   (coalesced over lanes)
//   B-op element (K=k,N=q) = Bcol[k*LDB + q]   (coalesced over lanes)
// A-frag layout (16x4 f32): lane l holds (M=l%16, K = 2*(l/16)+{0,1})
// B-frag layout (4x16 f32): lane l holds (K = 2*(l/16)+{0,1}, N=l%16)
// ------------------------------------------------------------------
template <int LDA, int LDB>
__device__ __forceinline__ void gemm_core_32x32(
    const float* __restrict__ Acol, const float* __restrict__ Bcol, int K,
    int mt, int nt, v8f acc[2][2]) {
  const int lane = threadIdx.x & 31;
  const int half = lane >> 4;
  const int lm = lane & 15;
  const float* ap0 = Acol + mt + lm;
  const float* bp0 = Bcol + nt + lm;

  int k0 = 0;
  const int Kmain = K - PF_ROWS;  // branch-free main loop w/ prefetch
#pragma unroll 4
  for (; k0 < Kmain; k0 += 4)
    gemm_step<LDA, LDB, true>(ap0, bp0, k0 + 2 * half, acc);
#pragma unroll
  for (; k0 < K; k0 += 4)  // tail: K,PF_ROWS const -> fully unrolled
    gemm_step<LDA, LDB, false>(ap0, bp0, k0 + 2 * half, acc);
}

// C-tile layout: lane l, vgpr v holds (M = v + 8*(l/16), N = l%16).

// ------------------------------------------------------------------
// LDS-tiled transpose (one-time): out[c*rows + r] = in[r*cols + c].
// 32x33 padded tile -> conflict-free banks; reads AND writes coalesced.
// rows, cols are multiples of 32 for all uses here.
// ------------------------------------------------------------------
__global__ void __launch_bounds__(256) transpose_tiled_kernel(
    const float* __restrict__ in, float* __restrict__ out, int rows,
    int cols) {
  __shared__ float tile[32][33];
  const int bx = blockIdx.x * 32;  // col base in `in`
  const int by = blockIdx.y * 32;  // row base in `in`
  const int tx = threadIdx.x;      // 0..31
  const int ty = threadIdx.y;      // 0..7
#pragma unroll
  for (int i = ty; i < 32; i += 8)
    tile[i][tx] = in[(size_t)(by + i) * cols + (bx + tx)];
  __syncthreads();
#pragma unroll
  for (int i = ty; i < 32; i += 8)
    out[(size_t)(bx + i) * rows + (by + tx)] = tile[tx][i];
}

// ------------------------------------------------------------------
// W^T builder: G = A^T A via WMMA (A is already K-major for both ops).
// G is symmetric, so the (p,q) tile stores directly (coalesced) into
//   WT[p*n+q] = W[q,p] = (1-m)*d(p,q) - G[p,q] + S[q,p] - S[p,q]
// ------------------------------------------------------------------
__global__ void __launch_bounds__(128) build_w_kernel(
    const float* __restrict__ A, const float* __restrict__ S,
    const float* __restrict__ m_raw, float* __restrict__ WT) {
  const int wave = threadIdx.x >> 5;
  const int mt = blockIdx.x * 128 + wave * 32;
  const int nt = blockIdx.y * 32;
  v8f acc[2][2];
  zero_acc(acc);
  gemm_core_32x32<N_DIM, N_DIM>(A, A, N_DIM, mt, nt, acc);

  const float mr = m_raw[0];
  const float m = (mr > 20.0f) ? mr : log1pf(expf(mr));  // softplus
  const float diag = 1.0f - m;
  const int lane = threadIdx.x & 31;
  const int half = lane >> 4;
  const int lm = lane & 15;
#pragma unroll
  for (int ti = 0; ti < 2; ++ti)
#pragma unroll
    for (int tj = 0; tj < 2; ++tj)
#pragma unroll
      for (int v = 0; v < 8; ++v) {
        const int p = mt + 16 * ti + v + 8 * half;
        const int q = nt + 16 * tj + lm;
        float w = ((p == q) ? diag : 0.0f) - acc[ti][tj][v] +
                  S[(size_t)q * N_DIM + p] - S[(size_t)p * N_DIM + q];
        WT[(size_t)p * N_DIM + q] = w;
      }
}

// ------------------------------------------------------------------
// Uxb = U x^T + b  (via UT, XT staged K-major), and fold iteration 1:
//   z1 = relu(0.5 * Uxb)    (since z0 = 0)
// ------------------------------------------------------------------
__global__ void __launch_bounds__(128) uxb_seed_kernel(
    const float* __restrict__ UT, const float* __restrict__ XT,
    const float* __restrict__ bvec, float* __restrict__ UXB,
    float* __restrict__ Z1) {
  const int wave = threadIdx.x >> 5;
  const int mt = blockIdx.x * 128 + wave * 32;
  const int nt = blockIdx.y * 32;
  v8f acc[2][2];
  zero_acc(acc);
  gemm_core_32x32<N_DIM, B_DIM>(UT, XT, D_DIM, mt, nt, acc);

  const int lane = threadIdx.x & 31;
  const int half = lane >> 4;
  const int lm = lane & 15;
#pragma unroll
  for (int ti = 0; ti < 2; ++ti)
#pragma unroll
    for (int tj = 0; tj < 2; ++tj)
#pragma unroll
      for (int v = 0; v < 8; ++v) {
        const int p = mt + 16 * ti + v + 8 * half;
        const int q = nt + 16 * tj + lm;
        const size_t idx = (size_t)p * B_DIM + q;
        float u = acc[ti][tj][v] + bvec[p];
        UXB[idx] = u;
        Z1[idx] = fmaxf(0.5f * u, 0.0f);
      }
}

// ------------------------------------------------------------------
// One FB iteration: zout = relu(0.5*(W zin + Uxb + zin)).
// Fused convergence check: every wave re-reduces the previous
// iteration's per-wave norm slots with a FIXED summation tree, so all
// waves/blocks reach a bit-identical decision (stream order guarantees
// the slots are complete). If converged: deterministic pass-through
// copy, and the slots are left untouched (sticky decision).
// ------------------------------------------------------------------
__global__ void __launch_bounds__(128) iterate_kernel(
    const float* __restrict__ WT, const float* __restrict__ UXB,
    const float* __restrict__ zin, float* __restrict__ zout,
    float* __restrict__ wave_sums) {
  const int wave = threadIdx.x >> 5;
  const int mt = blockIdx.x * 128 + wave * 32;
  const int nt = blockIdx.y * 32;
  const int lane = threadIdx.x & 31;
  const int half = lane >> 4;
  const int lm = lane & 15;

  // ---- fused convergence check (identical fixed-order in every wave) ----
  float d2p = 0.f, n2p = 0.f;
#pragma unroll
  for (int i = 0; i < NSLOTS / 32; ++i) {
    const int s = lane + 32 * i;
    d2p += wave_sums[2 * s];
    n2p += wave_sums[2 * s + 1];
  }
#pragma unroll
  for (int off = 16; off > 0; off >>= 1) {
    d2p += __shfl_xor(d2p, off, 32);
    n2p += __shfl_xor(n2p, off, 32);
  }
  // err = sqrt(d2)/(sqrt(n2)+1e-8) <= TOL  (same decision in all lanes)
  const bool converged = (sqrtf(d2p) <= TOL_F * (sqrtf(n2p) + 1e-8f));

  if (converged) {  // wave-uniform branch: pass-through copy
#pragma unroll
    for (int ti = 0; ti < 2; ++ti)
#pragma unroll
      for (int tj = 0; tj < 2; ++tj)
#pragma unroll
        for (int v = 0; v < 8; ++v) {
          const int p = mt + 16 * ti + v + 8 * half;
          const int q = nt + 16 * tj + lm;
          const size_t idx = (size_t)p * B_DIM + q;
          zout[idx] = zin[idx];
        }
    return;
  }

  v8f acc[2][2];
  zero_acc(acc);
  gemm_core_32x32<N_DIM, B_DIM>(WT, zin, N_DIM, mt, nt, acc);

  float d2 = 0.f, n2 = 0.f;
#pragma unroll
  for (int ti = 0; ti < 2; ++ti)
#pragma unroll
    for (int tj = 0; tj < 2; ++tj)
#pragma unroll
      for (int v = 0; v < 8; ++v) {
        const int p = mt + 16 * ti + v + 8 * half;
        const int q = nt + 16 * tj + lm;
        const size_t idx = (size_t)p * B_DIM + q;
        float zi = zin[idx];
        float zn = 0.5f * (acc[ti][tj][v] + UXB[idx] + zi);
        zn = fmaxf(zn, 0.0f);
        zout[idx] = zn;
        float df = zn - zi;
        d2 += df * df;
        n2 += zn * zn;
      }
  // wave32 butterfly reduce (deterministic order)
#pragma unroll
  for (int off = 16; off > 0; off >>= 1) {
    d2 += __shfl_xor(d2, off, 32);
    n2 += __shfl_xor(n2, off, 32);
  }
  if (lane == 0) {
    int slot = (blockIdx.y * gridDim.x + blockIdx.x) * 4 + wave;
    wave_sums[2 * slot] = d2;
    wave_sums[2 * slot + 1] = n2;
  }
}

// Seed the norm slots so iteration 1 always computes (d2=1, n2=0 ->
// err huge). Also makes graph replays independent of leftover ws state.
__global__ void __launch_bounds__(256) init_kernel(
    float* __restrict__ wave_sums) {
  int t = threadIdx.x;
  if (t < NSLOTS) {
    wave_sums[2 * t] = 1.0f;
    wave_sums[2 * t + 1] = 0.0f;
  }
}

// ------------------------------------------------------------------
extern "C" void kernel_launch(void* const* d_in, const int* in_sizes, int n_in,
                              void* d_out, int out_size, void* d_ws,
                              size_t ws_size, hipStream_t stream) {
  const float* A = (const float*)d_in[0];      // (1024,1024)
  const float* S = (const float*)d_in[1];      // (1024,1024)
  const float* m_raw = (const float*)d_in[2];  // scalar
  const float* U = (const float*)d_in[3];      // (1024,512)
  const float* bvec = (const float*)d_in[4];   // (1024,)
  const float* x = (const float*)d_in[5];      // (256,512)
  float* out = (float*)d_out;                  // (1024,256)

  float* ws = (float*)d_ws;
  size_t off = 0;
  float* WT = ws + off;        off += (size_t)N_DIM * N_DIM;  // 4 MB
  float* UXB = ws + off;       off += (size_t)N_DIM * B_DIM;  // 1 MB
  float* ZA = ws + off;        off += (size_t)N_DIM * B_DIM;
  float* ZB = ws + off;        off += (size_t)N_DIM * B_DIM;
  float* UT = ws + off;        off += (size_t)D_DIM * N_DIM;  // 2 MB
  float* XT = ws + off;        off += (size_t)D_DIM * B_DIM;  // 0.5 MB
  float* wave_sums = ws + off; off += 2 * NSLOTS;

  init_kernel<<<1, 256, 0, stream>>>(wave_sums);
  // UT[k*N + m] = U[m*D + k]
  transpose_tiled_kernel<<<dim3(D_DIM / 32, N_DIM / 32), dim3(32, 8), 0,
                           stream>>>(U, UT, N_DIM, D_DIM);
  // XT[k*B + c] = x[c*D + k]
  transpose_tiled_kernel<<<dim3(D_DIM / 32, B_DIM / 32), dim3(32, 8), 0,
                           stream>>>(x, XT, B_DIM, D_DIM);
  build_w_kernel<<<dim3(N_DIM / 128, N_DIM / 32), 128, 0, stream>>>(A, S,
                                                                    m_raw, WT);
  uxb_seed_kernel<<<dim3(N_DIM / 128, B_DIM / 32), 128, 0, stream>>>(
      UT, XT, bvec, UXB, ZA);

  // z1 done; run remaining 99 iterations (caps at z100 like MAX_ITERS).
  float* cur = ZA;
  float* nxt = ZB;
  const int remaining = 99;
  for (int it = 0; it < remaining; ++it) {
    float* dst = (it == remaining - 1) ? out : nxt;
    iterate_kernel<<<dim3(N_DIM / 128, B_DIM / 32), 128, 0, stream>>>(
        WT, UXB, cur, dst, wave_sums);
    float* t = cur;
    cur = dst;
    nxt = t;
  }
}